// MessagePassing_66786741453363
// MI455X (gfx1250) — compile-verified
//
#include <hip/hip_runtime.h>
#include <math.h>

// ---------------------------------------------------------------------------
// Graph attention message passing (gather -> attention matvec -> scatter-add)
// MI455X / gfx1250: async-to-LDS gather (ASYNCcnt) + V_WMMA_F32_16X16X4_F32
// for the per-edge 64-length attention dot products.
// ---------------------------------------------------------------------------

#define WAVES_PER_BLOCK 8
#define EDGES_PER_WAVE  16
#define ROW_DW          68   // 64 concat dwords + 4 pad: 272B row (=17*16B, keeps
                             // B128 alignment, kills LDS bank conflicts on A reads)

typedef float v2f __attribute__((ext_vector_type(2)));
typedef float v8f __attribute__((ext_vector_type(8)));

// LDS byte offset of a __shared__ object (addrspace(3) pointers ARE offsets
// from the wave's LDS base, which is what the async-load VDST operand wants).
__device__ __forceinline__ unsigned lds_off(const void* p) {
    return (unsigned)(unsigned long long)
        (__attribute__((address_space(3))) const void*)(p);
}

__global__ void zero_f32_kernel(float* __restrict__ p, int n) {
    int i = blockIdx.x * blockDim.x + threadIdx.x;
    if (i < n) p[i] = 0.0f;
}

__global__ void __launch_bounds__(256)
mp_attn_kernel(const float* __restrict__ xs,     // (N, 32)
               const float* __restrict__ nv,     // (E,)
               const float* __restrict__ aw,     // (64,)
               const int*   __restrict__ tIdx,   // (E,)
               const int*   __restrict__ sIdx,   // (E,)
               float*       __restrict__ out,    // (N, 32), pre-zeroed
               int E)
{
    __shared__ __align__(16) float sRows[WAVES_PER_BLOCK][EDGES_PER_WAVE * ROW_DW];
    __shared__ float sW[64];
    __shared__ float sDots[WAVES_PER_BLOCK][EDGES_PER_WAVE];

    const int tid  = threadIdx.x;
    const int wv   = tid >> 5;     // wave id in block (wave32)
    const int lane = tid & 31;

    if (tid < 64) sW[tid] = aw[tid];
    __syncthreads();

    const int tile = blockIdx.x * WAVES_PER_BLOCK + wv;
    const int Eb   = tile * EDGES_PER_WAVE;
    if (Eb >= E) return;           // wave-uniform exit (after the only barrier)

    // ---- Stage 1: async gather of 16 edges x {src,tgt} rows into LDS -------
    // lane -> (edge slot = lane&15, half: 0=msg/src row, 1=tgt row)
    const int slot = lane & 15;
    const int half = lane >> 4;
    int eg = Eb + slot; if (eg >= E) eg = E - 1;        // clamp keeps EXEC full
    const int node = (half ? tIdx : sIdx)[eg];
    float*       rowDst = &sRows[wv][slot * ROW_DW + half * 32];
    const float* gSrc   = xs + (size_t)node * 32;
#pragma unroll
    for (int p = 0; p < 8; ++p) {                       // 8 x 16B per 128B row
        unsigned loff = lds_off(rowDst + p * 4);
        const float* gp = gSrc + p * 4;
        asm volatile("global_load_async_to_lds_b128 %0, %1, off"
                     :: "v"(loff), "v"(gp) : "memory");
    }
    asm volatile("s_wait_asynccnt 0" ::: "memory");
    __builtin_amdgcn_wave_barrier();

    // ---- Stage 2: attention dots via V_WMMA_F32_16X16X4_F32 ----------------
    // A (16x4 f32): M = lane&15 (edge), K pair = 2*(lane>>4) + {0,1} per step.
    // B (4x16 f32): broadcast w over N; per-K lane mapping mirrors A:
    //   VGPR0 = {k=0 | lanes 0-15, k=2 | lanes 16-31}, VGPR1 = {k=1, k=3}.
    // D[m,n] = dot(concat_row(m), w) for every n.
    const float* myRow = &sRows[wv][slot * ROW_DW];
    const int koff = half * 2;
    v8f acc = {};
#pragma unroll
    for (int s = 0; s < 16; ++s) {                      // K = 64 in steps of 4
        v2f a, b;
        a.x = myRow[4 * s + koff];
        a.y = myRow[4 * s + koff + 1];
        b.x = sW[4 * s + koff];
        b.y = sW[4 * s + koff + 1];
        acc = __builtin_amdgcn_wmma_f32_16x16x4_f32(
                  false, a, false, b, (short)0, acc, false, false);
    }

    // Column n=0 of D lives in lanes 0 (edges 0..7) and 16 (edges 8..15).
    if ((lane & 15) == 0) {
        const int base = half * 8;
#pragma unroll
        for (int r = 0; r < 8; ++r) sDots[wv][base + r] = acc[r];
    }
    __builtin_amdgcn_wave_barrier();   // same-wave DS ops are in-order in HW

    // ---- Stage 3: ELU, scale, scatter-add (2 lanes x 16 channels per edge) -
    const int e2 = Eb + slot;
    if (e2 < E) {
        float d   = sDots[wv][slot];
        float att = d > 0.0f ? d : expm1f(d);            // ELU(alpha=1)
        float val = nv[e2] * att;
        int   t   = tIdx[e2];
        float*       op = out + (size_t)t * 32 + half * 16;
        const float* mp = &sRows[wv][slot * ROW_DW + half * 16]; // msg row half
#pragma unroll
        for (int j = 0; j < 16; ++j)
            atomicAdd(op + j, val * mp[j]);              // global_atomic_add_f32
    }
}

extern "C" void kernel_launch(void* const* d_in, const int* in_sizes, int n_in,
                              void* d_out, int out_size, void* d_ws, size_t ws_size,
                              hipStream_t stream) {
    const float* xs   = (const float*)d_in[0];   // x_source (N,32)
    const float* nv   = (const float*)d_in[1];   // neighborhood_values (E,)
    const float* aw   = (const float*)d_in[2];   // att_weight (64,)
    const int*   tIdx = (const int*)d_in[3];     // target_index (E,)
    const int*   sIdx = (const int*)d_in[4];     // source_index (E,)
    float*       out  = (float*)d_out;           // (n_target, 32)
    const int E = in_sizes[1];

    int zb = (out_size + 255) / 256;
    if (zb > 0) zero_f32_kernel<<<zb, 256, 0, stream>>>(out, out_size);

    int tiles  = (E + EDGES_PER_WAVE - 1) / EDGES_PER_WAVE;
    int blocks = (tiles + WAVES_PER_BLOCK - 1) / WAVES_PER_BLOCK;
    if (blocks > 0)
        mp_attn_kernel<<<blocks, 256, 0, stream>>>(xs, nv, aw, tIdx, sIdx, out, E);
}